// GraphAttn_24154896073110
// MI455X (gfx1250) — compile-verified
//
#include <hip/hip_runtime.h>
#include <hip/hip_bf16.h>

#define NN 50000
#define EE 800000
#define IN_DIM 256
#define HD 128            // H*D
#define NHEAD 4
#define DHEAD 32
#define NEG_SLOPE 0.2f
#define BN_EPS 1e-5f

typedef float v2f __attribute__((ext_vector_type(2)));
typedef float v8f __attribute__((ext_vector_type(8)));

// ---- order-preserving float<->int mapping for atomic max on signed int ----
__device__ __forceinline__ int f2o(float f) {
    int i = __float_as_int(f);
    return i >= 0 ? i : (i ^ 0x7FFFFFFF);
}
__device__ __forceinline__ float o2f(int o) {
    int i = o >= 0 ? o : (o ^ 0x7FFFFFFF);
    return __int_as_float(i);
}

// ---------------- init workspace (harness poisons ws with 0xAA) ----------------
__global__ void init_ws(int* __restrict__ mord, float* __restrict__ sden,
                        float* __restrict__ rst, float* __restrict__ ssum,
                        float* __restrict__ ssumq) {
    int tid = blockIdx.x * blockDim.x + threadIdx.x;
    int stride = gridDim.x * blockDim.x;
    for (int i = tid; i < NN * HD; i += stride) rst[i] = 0.0f;
    for (int i = tid; i < NN * NHEAD; i += stride) {
        sden[i] = 0.0f;
        mord[i] = f2o(-1e30f);
    }
    if (tid < HD) { ssum[tid] = 0.0f; ssumq[tid] = 0.0f; }
}

// ---------------- h = x @ W via FP32 WMMA 16x16x4 ----------------
// block: 256 threads = 8 waves. wave w -> output cols [16w,16w+16), rows [16*bid,...)
__global__ __launch_bounds__(256) void gat_gemm_wmma(const float* __restrict__ x,
                                                     const float* __restrict__ W,
                                                     float* __restrict__ hbuf) {
    const int wave = threadIdx.x >> 5;   // 0..7
    const int lane = threadIdx.x & 31;
    const int m0 = blockIdx.x * 16;
    const int n0 = wave * 16;
    const int l16 = lane & 15;
    const int khalf = (lane >> 4) * 2;   // 0 or 2

    v8f c = {};
    const float* xrow = x + (size_t)(m0 + l16) * IN_DIM;

    for (int k0 = 0; k0 < IN_DIM; k0 += 4) {
        v2f a, b;
        // A 16x4: lanes0-15 hold K=khalf+{0,1} of row M=l16 (khalf=0), lanes16-31 khalf=2
        a.x = xrow[k0 + khalf];
        a.y = xrow[k0 + khalf + 1];
        // B 4x16: lane half selects K pair, l16 = N
        b.x = W[(size_t)(k0 + khalf) * HD + n0 + l16];
        b.y = W[(size_t)(k0 + khalf + 1) * HD + n0 + l16];
        c = __builtin_amdgcn_wmma_f32_16x16x4_f32(false, a, false, b,
                                                  (short)0, c, false, false);
    }
    // C/D 16x16 f32: VGPR r -> M = r + 8*(lane/16), N = lane%16
    const int mbase = m0 + (lane >> 4) * 8;
    const int col = n0 + l16;
#pragma unroll
    for (int r = 0; r < 8; ++r)
        hbuf[(size_t)(mbase + r) * HD + col] = c[r];
}

// ---------------- el/er per (node, head) ----------------
__global__ void compute_elr(const float* __restrict__ hbuf,
                            const float* __restrict__ attn_l,
                            const float* __restrict__ attn_r,
                            float* __restrict__ el, float* __restrict__ er) {
    int i = blockIdx.x * blockDim.x + threadIdx.x;   // n*NHEAD + head
    if (i >= NN * NHEAD) return;
    int n = i >> 2, hd = i & 3;
    const float* hp = hbuf + (size_t)n * HD + hd * DHEAD;
    const float* al = attn_l + hd * DHEAD;
    const float* ar = attn_r + hd * DHEAD;
    float sl = 0.f, sr = 0.f;
#pragma unroll
    for (int d = 0; d < DHEAD; ++d) {
        float v = hp[d];
        sl += v * al[d];
        sr += v * ar[d];
    }
    el[i] = sl;
    er[i] = sr;
}

// ---------------- edge pass A: leaky_relu(el[src]+er[dst]) + segment max ----------------
__global__ void edge_max(const int* __restrict__ src, const int* __restrict__ dst,
                         const float* __restrict__ el, const float* __restrict__ er,
                         float* __restrict__ ebuf, int* __restrict__ mord) {
    int i = blockIdx.x * blockDim.x + threadIdx.x;
    if (i >= EE * NHEAD) return;
    int e = i >> 2, hd = i & 3;
    int s = src[e], d = dst[e];
    float v = el[s * NHEAD + hd] + er[d * NHEAD + hd];
    v = v > 0.f ? v : NEG_SLOPE * v;
    ebuf[i] = v;
    atomicMax(&mord[d * NHEAD + hd], f2o(v));   // global_atomic_max_i32 in L2
}

// ---------------- edge pass B: exp(e - m[dst]) + segment sum ----------------
__global__ void edge_exp(const int* __restrict__ dst, const int* __restrict__ mord,
                         float* __restrict__ ebuf, float* __restrict__ sden) {
    int i = blockIdx.x * blockDim.x + threadIdx.x;
    if (i >= EE * NHEAD) return;
    int e = i >> 2, hd = i & 3;
    int d = dst[e];
    float ex = __expf(ebuf[i] - o2f(mord[d * NHEAD + hd]));
    ebuf[i] = ex;
    atomicAdd(&sden[d * NHEAD + hd], ex);       // global_atomic_add_f32 in L2
}

// ---------------- edge pass C: rst[dst] += h[src] * alpha ----------------
// one block per edge, 128 threads = full feature vector (coalesced 512B gather of h[src])
__global__ __launch_bounds__(128) void edge_aggregate(const int* __restrict__ src,
                                                      const int* __restrict__ dst,
                                                      const float* __restrict__ hbuf,
                                                      const float* __restrict__ ebuf,
                                                      const float* __restrict__ sden,
                                                      float* __restrict__ rst) {
    int e = blockIdx.x;
    int i = threadIdx.x;     // 0..127
    int hd = i >> 5;         // head
    int s = src[e], d = dst[e];
    float alpha = ebuf[e * NHEAD + hd] / sden[d * NHEAD + hd];
    atomicAdd(&rst[(size_t)d * HD + i], hbuf[(size_t)s * HD + i] * alpha);
}

// ---------------- BN stats: per-feature partial sums ----------------
__global__ __launch_bounds__(256) void bn_partial(const float* __restrict__ rst,
                                                  const float* __restrict__ bias,
                                                  float* __restrict__ ssum,
                                                  float* __restrict__ ssumq) {
    int j = threadIdx.x & (HD - 1);      // feature
    int sub = threadIdx.x >> 7;          // 0/1
    float b = bias[j];
    float s = 0.f, q = 0.f;
    for (int n = blockIdx.x * 2 + sub; n < NN; n += gridDim.x * 2) {
        float v = rst[(size_t)n * HD + j] + b;
        s += v;
        q += v * v;
    }
    atomicAdd(&ssum[j], s);
    atomicAdd(&ssumq[j], q);
}

__global__ void bn_finalize(const float* __restrict__ ssum, const float* __restrict__ ssumq,
                            float* __restrict__ mean, float* __restrict__ rstd) {
    int j = threadIdx.x;                 // 128 threads
    float m = ssum[j] * (1.0f / NN);
    float v = ssumq[j] * (1.0f / NN) - m * m;
    mean[j] = m;
    rstd[j] = rsqrtf(v + BN_EPS);
}

// ---------------- apply BN + ReLU ----------------
__global__ void bn_apply(const float* __restrict__ rst, const float* __restrict__ bias,
                         const float* __restrict__ gamma, const float* __restrict__ beta,
                         const float* __restrict__ mean, const float* __restrict__ rstd,
                         float* __restrict__ out) {
    int i = blockIdx.x * blockDim.x + threadIdx.x;
    if (i >= NN * HD) return;
    int j = i & (HD - 1);
    float y = gamma[j] * (rst[i] + bias[j] - mean[j]) * rstd[j] + beta[j];
    out[i] = fmaxf(y, 0.0f);
}

extern "C" void kernel_launch(void* const* d_in, const int* in_sizes, int n_in,
                              void* d_out, int out_size, void* d_ws, size_t ws_size,
                              hipStream_t stream) {
    const float* x      = (const float*)d_in[0];
    const int*   src    = (const int*)d_in[1];
    const int*   dst    = (const int*)d_in[2];
    const float* W      = (const float*)d_in[3];
    const float* attn_l = (const float*)d_in[4];
    const float* attn_r = (const float*)d_in[5];
    const float* bias   = (const float*)d_in[6];
    const float* gamma  = (const float*)d_in[7];
    const float* beta   = (const float*)d_in[8];
    float* out = (float*)d_out;

    // workspace layout (all chunks 256B-aligned by construction)
    char* ws = (char*)d_ws;
    float* hbuf = (float*)ws;  ws += (size_t)NN * HD * 4;       // 25.6 MB
    float* el   = (float*)ws;  ws += (size_t)NN * NHEAD * 4;    // 0.8 MB
    float* er   = (float*)ws;  ws += (size_t)NN * NHEAD * 4;    // 0.8 MB
    float* ebuf = (float*)ws;  ws += (size_t)EE * NHEAD * 4;    // 12.8 MB
    int*   mord = (int*)ws;    ws += (size_t)NN * NHEAD * 4;    // 0.8 MB
    float* sden = (float*)ws;  ws += (size_t)NN * NHEAD * 4;    // 0.8 MB
    float* rst  = (float*)ws;  ws += (size_t)NN * HD * 4;       // 25.6 MB
    float* ssum = (float*)ws;  ws += 512;
    float* ssumq= (float*)ws;  ws += 512;
    float* mean = (float*)ws;  ws += 512;
    float* rstd = (float*)ws;  ws += 512;

    init_ws<<<4096, 256, 0, stream>>>(mord, sden, rst, ssum, ssumq);
    gat_gemm_wmma<<<NN / 16, 256, 0, stream>>>(x, W, hbuf);
    compute_elr<<<(NN * NHEAD + 255) / 256, 256, 0, stream>>>(hbuf, attn_l, attn_r, el, er);
    edge_max<<<(EE * NHEAD + 255) / 256, 256, 0, stream>>>(src, dst, el, er, ebuf, mord);
    edge_exp<<<(EE * NHEAD + 255) / 256, 256, 0, stream>>>(dst, mord, ebuf, sden);
    edge_aggregate<<<EE, 128, 0, stream>>>(src, dst, hbuf, ebuf, sden, rst);
    bn_partial<<<256, 256, 0, stream>>>(rst, bias, ssum, ssumq);
    bn_finalize<<<1, HD, 0, stream>>>(ssum, ssumq, mean, rstd);
    bn_apply<<<(NN * HD + 255) / 256, 256, 0, stream>>>(rst, bias, gamma, beta, mean, rstd, out);
}